// TopKGate_71330816852132
// MI455X (gfx1250) — compile-verified
//
#include <hip/hip_runtime.h>
#include <hip/hip_bf16.h>
#include <math.h>

typedef __attribute__((ext_vector_type(2))) float v2f;
typedef __attribute__((ext_vector_type(4))) float f4;
typedef __attribute__((ext_vector_type(8))) float v8f;

#define T_TOK 16384
#define DMODEL 2048
#define NEXP 64
#define BM 64       // tokens per workgroup (4 waves x 16 rows)
#define KC 32       // k-chunk depth
#define XS 36       // x-tile LDS row stride (dwords): 36=4*9 -> conflict-free b64 frags
#define WTS 36      // transposed Wg-tile [n][k] stride (dwords): same property
#define LSTR 65     // logit/prob LDS row stride (dwords) -> conflict-free row scans
#define NCHUNK (DMODEL / KC)

#define XT_DW (BM * XS)      // 2304 dwords per x buffer
#define WT_DW (NEXP * WTS)   // 2304 dwords per wg buffer
#define AOPS  20             // async ops issued per thread per chunk (4 b128 + 16 b32)

__device__ __forceinline__ unsigned lds_addr(const float* p)
{
    return (unsigned)(uintptr_t)p;   // shared aperture: LDS offset = addr[31:0]
}

__device__ __forceinline__ void issue_chunk(const float* __restrict__ x,
                                            const float* __restrict__ wg,
                                            int row0, int k0,
                                            float* xtb, float* wtb, int tid)
{
    // x tile [BM x KC] : 512 f4 / 128 threads = 4 async b128 per thread
    #pragma unroll
    for (int j = 0; j < 4; ++j) {
        int q  = tid + j * 128;
        int r  = q >> 3;          // 8 f4 per row (KC=32)
        int c4 = q & 7;
        unsigned voff = (unsigned)(((row0 + r) * DMODEL + k0 + c4 * 4) * 4);
        unsigned lds  = lds_addr(&xtb[r * XS + c4 * 4]);
        asm volatile("global_load_async_to_lds_b128 %0, %1, %2 offset:0"
                     :: "v"(lds), "v"(voff), "s"(x) : "memory");
    }
    // Wg chunk [KC x 64] transposed into wtT[e][k]: async engine does the
    // scatter (coalesced 4B global reads, per-lane LDS destinations)
    #pragma unroll
    for (int j = 0; j < 16; ++j) {
        int q = tid + j * 128;
        int k = q >> 6;           // 64 floats per Wg row
        int e = q & 63;
        unsigned voff = (unsigned)(((k0 + k) * NEXP + e) * 4);
        unsigned lds  = lds_addr(&wtb[e * WTS + k]);
        asm volatile("global_load_async_to_lds_b32 %0, %1, %2 offset:0"
                     :: "v"(lds), "v"(voff), "s"(wg) : "memory");
    }
}

__device__ __forceinline__ void compute_chunk(const float* xtb, const float* wtb,
                                              int wv, int l16, int half,
                                              v8f& a0, v8f& a1, v8f& a2, v8f& a3)
{
    // A frag: lane<16 -> K={kk,kk+1}; lane>=16 -> K={kk+2,kk+3} (per-lane b64)
    const float* arow = &xtb[(wv * 16 + l16) * XS + half * 2];
    // B frag from transposed tile: per-lane b64 at wtT[n][kk+2h]
    const float* brow = &wtb[l16 * WTS + half * 2];
    #pragma unroll
    for (int kk = 0; kk < KC; kk += 4) {
        v2f a  = *(const v2f*)(arow + kk);
        v2f b0 = *(const v2f*)(brow +  0 * WTS + kk);
        v2f b1 = *(const v2f*)(brow + 16 * WTS + kk);
        v2f b2 = *(const v2f*)(brow + 32 * WTS + kk);
        v2f b3 = *(const v2f*)(brow + 48 * WTS + kk);
        a0 = __builtin_amdgcn_wmma_f32_16x16x4_f32(false, a, false, b0, (short)0, a0, false, false);
        a1 = __builtin_amdgcn_wmma_f32_16x16x4_f32(false, a, false, b1, (short)0, a1, false, false);
        a2 = __builtin_amdgcn_wmma_f32_16x16x4_f32(false, a, false, b2, (short)0, a2, false, false);
        a3 = __builtin_amdgcn_wmma_f32_16x16x4_f32(false, a, false, b3, (short)0, a3, false, false);
    }
}

__global__ __launch_bounds__(128)
void topk_gate_main(const float* __restrict__ x, const float* __restrict__ wg,
                    float* __restrict__ out, float* __restrict__ ws)
{
    // [ xt0 | xt1 | wt0 | wt1 | cnt ]  = 9280 dwords = 37120 B
    __shared__ __align__(16) float smem[2 * XT_DW + 2 * WT_DW + NEXP];
    float* xbase = smem;
    float* wbase = smem + 2 * XT_DW;
    float* cnt   = smem + 2 * XT_DW + 2 * WT_DW;

    const int tid  = threadIdx.x;
    const int blk  = blockIdx.x;
    const int row0 = blk * BM;

    // ---- zero this block's dense gate region (coalesced float4) ----
    {
        f4 z = {0.f, 0.f, 0.f, 0.f};
        f4* og = (f4*)(out + (size_t)row0 * NEXP);
        #pragma unroll
        for (int i = 0; i < (BM * NEXP / 4) / 128; ++i)
            og[tid + i * 128] = z;
    }
    if (tid < NEXP) cnt[tid] = 0.f;

    const int lane = tid & 31;
    const int wv   = tid >> 5;
    const int l16  = lane & 15;
    const int half = lane >> 4;

    v8f a0 = {}, a1 = {}, a2 = {}, a3 = {};

    // ---- double-buffered async pipeline over K ----
    issue_chunk(x, wg, row0, 0, xbase, wbase, tid);
    for (int i = 0; i < NCHUNK; ++i) {
        int cur = i & 1;
        if (i + 1 < NCHUNK) {
            issue_chunk(x, wg, row0, (i + 1) * KC,
                        xbase + (1 - cur) * XT_DW, wbase + (1 - cur) * WT_DW, tid);
            asm volatile("s_wait_asynccnt 20" ::: "memory");   // chunk i resident
        } else {
            asm volatile("s_wait_asynccnt 0" ::: "memory");
        }
        __syncthreads();
        compute_chunk(xbase + cur * XT_DW, wbase + cur * WT_DW,
                      wv, l16, half, a0, a1, a2, a3);
        __syncthreads();   // reads done before buffer is overwritten next round
    }

    // ---- logits -> LDS (stride 65). C layout: VGPR j = row j / row 8+j ----
    {
        int rbase = wv * 16 + half * 8;
        #pragma unroll
        for (int j = 0; j < 8; ++j) {
            smem[(rbase + j) * LSTR +  0 + l16] = a0[j];
            smem[(rbase + j) * LSTR + 16 + l16] = a1[j];
            smem[(rbase + j) * LSTR + 32 + l16] = a2[j];
            smem[(rbase + j) * LSTR + 48 + l16] = a3[j];
        }
    }
    __syncthreads();

    // ---- per-token softmax + top-2 + scatter (one thread per token) ----
    if (tid < BM) {
        float* L = &smem[tid * LSTR];
        float m = -__builtin_inff();
        float v1 = -__builtin_inff(), v2 = -__builtin_inff();
        int   i1 = 0, i2 = 0;
        for (int e = 0; e < NEXP; ++e) {
            float v = L[e];
            m = fmaxf(m, v);
            if (v > v1)      { v2 = v1; i2 = i1; v1 = v; i1 = e; }
            else if (v > v2) { v2 = v;  i2 = e; }
        }
        float s = 0.f;
        for (int e = 0; e < NEXP; ++e) { float ev = expf(L[e] - m); s += ev; L[e] = ev; }
        float w1 = expf(v1 - m), w2 = expf(v2 - m);
        float gs = 1.f / (w1 + w2);
        float rs = 1.f / s;
        for (int e = 0; e < NEXP; ++e) L[e] *= rs;   // probs (for P column sums)

        size_t grow = (size_t)(row0 + tid);
        out[grow * NEXP + i1] = w1 * gs;
        out[grow * NEXP + i2] = w2 * gs;
        float* oidx = out + (size_t)T_TOK * NEXP;
        oidx[grow * 2 + 0] = (float)i1;
        oidx[grow * 2 + 1] = (float)i2;
        atomicAdd(&cnt[i1], 1.f);
        atomicAdd(&cnt[i2], 1.f);
    }
    __syncthreads();

    // ---- per-expert partial sums -> global workspace ----
    if (tid < NEXP) {
        float s = 0.f;
        for (int r = 0; r < BM; ++r) s += smem[r * LSTR + tid];
        atomicAdd(&ws[tid], s);               // sum of probs per expert
        atomicAdd(&ws[NEXP + tid], cnt[tid]); // routing counts per expert
    }
}

__global__ void topk_gate_zero_ws(float* ws)
{
    if (threadIdx.x < 2 * NEXP) ws[threadIdx.x] = 0.f;
}

__global__ void topk_gate_finalize(const float* __restrict__ ws, float* __restrict__ out)
{
    __shared__ float red[NEXP];
    int e = threadIdx.x;
    red[e] = ws[e] * ws[NEXP + e];   // P_sum_e * cnt_e
    __syncthreads();
    if (e == 0) {
        float s = 0.f;
        for (int i = 0; i < NEXP; ++i) s += red[i];
        out[(size_t)T_TOK * NEXP + (size_t)T_TOK * 2] =
            (float)NEXP * s / ((float)T_TOK * (float)T_TOK);
    }
}

extern "C" void kernel_launch(void* const* d_in, const int* in_sizes, int n_in,
                              void* d_out, int out_size, void* d_ws, size_t ws_size,
                              hipStream_t stream)
{
    const float* x  = (const float*)d_in[0];   // [16384, 2048]
    const float* wg = (const float*)d_in[1];   // [2048, 64]
    float* out = (float*)d_out;                // gate[16384*64] | idx-as-f32[16384*2] | aux[1]
    float* ws  = (float*)d_ws;                 // [64 P-sums | 64 counts]

    topk_gate_zero_ws<<<1, 128, 0, stream>>>(ws);
    topk_gate_main<<<T_TOK / BM, 128, 0, stream>>>(x, wg, out, ws);
    topk_gate_finalize<<<1, NEXP, 0, stream>>>(ws, out);
}